// Process_3D_Motion_57612691309027
// MI455X (gfx1250) — compile-verified
//
#include <hip/hip_runtime.h>
#include <stdint.h>

// ---------------------------------------------------------------------------
// Forward kinematics (batch_rigid_transform), 55-joint tree, 32*2048 items.
// Memory-bound (~173 MB moved, ~0.34 GFLOP): stage input per-block into LDS
// via CDNA5 async global->LDS copies (coalesced), compute chain per lane from
// LDS (conflict-free odd strides), async-store positions LDS->global.
// ---------------------------------------------------------------------------

#define N_JOINTS 55
#define THREADS 64
#define ITEMS_PER_BLOCK 64
#define FLOATS_PER_ITEM (N_JOINTS * 9)                      // 495
#define MAT_REGION_FLOATS (ITEMS_PER_BLOCK * FLOATS_PER_ITEM) // 31680
#define POS_FLOATS_PER_ITEM (N_JOINTS * 3)                  // 165
#define POS_REGION_FLOATS (ITEMS_PER_BLOCK * POS_FLOATS_PER_ITEM) // 10560
#define REL_REGION_FLOATS 168                               // 55*3 padded

__constant__ int kParents[N_JOINTS] = {
    -1, 0, 0, 0, 1, 2, 3, 4, 5, 6, 7, 8, 9, 9, 9, 12, 13, 14, 16, 17,
    18, 19, 15, 22, 23, 20, 25, 26, 20, 28, 29, 20, 31, 32, 20, 34, 35,
    20, 37, 38, 21, 40, 41, 21, 43, 44, 21, 46, 47, 21, 49, 50, 21, 52, 53};

#if defined(__has_builtin)
#if __has_builtin(__builtin_amdgcn_global_load_async_to_lds_b128)
#define HAVE_ASYNC_LOAD 1
#endif
#if __has_builtin(__builtin_amdgcn_global_store_async_from_lds_b128)
#define HAVE_ASYNC_STORE 1
#endif
#if __has_builtin(__builtin_amdgcn_s_wait_asynccnt)
#define HAVE_ASYNC_WAIT 1
#endif
#endif

// Builtin parameter types (from the compiler diagnostic): pointers to a
// 16-byte int vector, in addrspace(1) (global / __device__) and addrspace(3)
// (LDS / __shared__) respectively.
typedef int v4i __attribute__((vector_size(4 * sizeof(int))));
typedef __attribute__((address_space(1))) v4i* gptr4;
typedef __attribute__((address_space(3))) v4i* lptr4;

__device__ static inline void async_g2l_b128(const float* g, float* l) {
#if defined(HAVE_ASYNC_LOAD)
  // Low 32 bits of a flat shared pointer are the LDS byte offset (aperture
  // mapping).
  __builtin_amdgcn_global_load_async_to_lds_b128(
      (gptr4)(uintptr_t)g, (lptr4)(uint32_t)(uintptr_t)l, 0, 0);
#else
  *(float4*)l = *(const float4*)g;
#endif
}

__device__ static inline void async_l2g_b128(const float* l, float* g) {
#if defined(HAVE_ASYNC_STORE)
  __builtin_amdgcn_global_store_async_from_lds_b128(
      (gptr4)(uintptr_t)g, (lptr4)(uint32_t)(uintptr_t)l, 0, 0);
#else
  *(float4*)g = *(const float4*)l;
#endif
}

__device__ static inline void wait_async0() {
#if defined(HAVE_ASYNC_WAIT)
  __builtin_amdgcn_s_wait_asynccnt(0);
#elif defined(HAVE_ASYNC_LOAD) || defined(HAVE_ASYNC_STORE)
  asm volatile("s_wait_asynccnt 0" ::: "memory");
#endif
}

__global__ __launch_bounds__(THREADS) void fk_chain_kernel(
    const float* __restrict__ motion, const float* __restrict__ J,
    float* __restrict__ out) {
  extern __shared__ float lds[];
  float* mats = lds;                                 // [64][495] (R, then chain rot in-place)
  float* pos = lds + MAT_REGION_FLOATS;              // [64][165]
  float* rel = pos + POS_REGION_FLOATS;              // [55][3]

  const int tid = threadIdx.x;
  const long long itemBase = (long long)blockIdx.x * ITEMS_PER_BLOCK;
  const float* gmat = motion + itemBase * FLOATS_PER_ITEM;

  // ---- Phase 1: coalesced async stage of this block's rotation matrices ----
  const int nVec4 = MAT_REGION_FLOATS / 4;  // 7920 x 16B
  for (int v = tid; v < nVec4; v += THREADS) {
    async_g2l_b128(gmat + v * 4, mats + v * 4);
  }

  // rel offsets (tiny, broadcast reads of J)
  if (tid < N_JOINTS) {
    const int p = kParents[tid];
    float x = J[tid * 3 + 0], y = J[tid * 3 + 1], z = J[tid * 3 + 2];
    if (p >= 0) {
      x -= J[p * 3 + 0];
      y -= J[p * 3 + 1];
      z -= J[p * 3 + 2];
    }
    rel[tid * 3 + 0] = x;
    rel[tid * 3 + 1] = y;
    rel[tid * 3 + 2] = z;
  }

  wait_async0();
  __syncthreads();

  // ---- Phase 2: each lane walks its item's kinematic chain from LDS ----
  float* m = mats + tid * FLOATS_PER_ITEM;     // lane stride 495 (odd) -> no bank conflicts
  float* ps = pos + tid * POS_FLOATS_PER_ITEM; // lane stride 165 (odd) -> no bank conflicts

  // joint 0: chain rot == R0 already in place; pos = rel[0] (= J[0])
  ps[0] = rel[0];
  ps[1] = rel[1];
  ps[2] = rel[2];

  for (int j = 1; j < N_JOINTS; ++j) {
    const int p = kParents[j];  // uniform per wave -> scalar load
    const float* Pm = m + p * 9;
    float* Rm = m + j * 9;
    const float p00 = Pm[0], p01 = Pm[1], p02 = Pm[2];
    const float p10 = Pm[3], p11 = Pm[4], p12 = Pm[5];
    const float p20 = Pm[6], p21 = Pm[7], p22 = Pm[8];
    const float r00 = Rm[0], r01 = Rm[1], r02 = Rm[2];
    const float r10 = Rm[3], r11 = Rm[4], r12 = Rm[5];
    const float r20 = Rm[6], r21 = Rm[7], r22 = Rm[8];
    // chain rot = P * R, written back over the input slot
    Rm[0] = p00 * r00 + p01 * r10 + p02 * r20;
    Rm[1] = p00 * r01 + p01 * r11 + p02 * r21;
    Rm[2] = p00 * r02 + p01 * r12 + p02 * r22;
    Rm[3] = p10 * r00 + p11 * r10 + p12 * r20;
    Rm[4] = p10 * r01 + p11 * r11 + p12 * r21;
    Rm[5] = p10 * r02 + p11 * r12 + p12 * r22;
    Rm[6] = p20 * r00 + p21 * r10 + p22 * r20;
    Rm[7] = p20 * r01 + p21 * r11 + p22 * r21;
    Rm[8] = p20 * r02 + p21 * r12 + p22 * r22;
    // chain pos = P * rel_j + parent pos
    const float rx = rel[j * 3 + 0], ry = rel[j * 3 + 1], rz = rel[j * 3 + 2];
    ps[j * 3 + 0] = p00 * rx + p01 * ry + p02 * rz + ps[p * 3 + 0];
    ps[j * 3 + 1] = p10 * rx + p11 * ry + p12 * rz + ps[p * 3 + 1];
    ps[j * 3 + 2] = p20 * rx + p21 * ry + p22 * rz + ps[p * 3 + 2];
  }

  __syncthreads();

  // ---- Phase 3: coalesced async store of positions LDS -> global ----
  float* gout = out + itemBase * POS_FLOATS_PER_ITEM;
  const int nOut4 = POS_REGION_FLOATS / 4;  // 2640 x 16B
  for (int v = tid; v < nOut4; v += THREADS) {
    async_l2g_b128(pos + v * 4, gout + v * 4);
  }
  wait_async0();  // drain before wave exit (S_ENDPGM also implies wait-idle)
}

extern "C" void kernel_launch(void* const* d_in, const int* in_sizes, int n_in,
                              void* d_out, int out_size, void* d_ws,
                              size_t ws_size, hipStream_t stream) {
  const float* motion = (const float*)d_in[0];
  const float* J = (const float*)d_in[1];
  float* out = (float*)d_out;

  const int items = in_sizes[0] / FLOATS_PER_ITEM;  // 65536
  const int blocks = items / ITEMS_PER_BLOCK;       // 1024
  const size_t shmem =
      (size_t)(MAT_REGION_FLOATS + POS_REGION_FLOATS + REL_REGION_FLOATS) *
      sizeof(float);  // ~169.6 KB of the 320 KB WGP LDS

  fk_chain_kernel<<<blocks, THREADS, shmem, stream>>>(motion, J, out);
}